// DISCOBlock_57801669869705
// MI455X (gfx1250) — compile-verified
//
#include <hip/hip_runtime.h>
#include <math.h>

// Problem constants
#define NLAT 91
#define NLON 180
#define NFREQ 91        // rfft bins
#define CIN 64
#define COUT 64
#define KSZ 25
#define NB 2
#define WWIN 9          // 2*4+1
#define BC (NB*CIN)     // 128
#define KDIM (CIN*KSZ)  // 1600
#define NCOLS (NB*NLAT*NFREQ*2)   // 33124 (re/im as separate columns)
#define NPAD 33136                 // padded to multiple of 16
#define NTILES (NPAD/16)           // 2071
#define KCHUNK (KDIM/8)            // 200 K per wave (8 waves/block)
#define KSTEPS (KCHUNK/4)          // 50 k-steps per wave
#define DEPTH 4                    // async staging depth (power of two)

typedef float v2f __attribute__((ext_vector_type(2)));
typedef float v8f __attribute__((ext_vector_type(8)));

// ---------------------------------------------------------------------------
// Kernel 1: Xf = rfft(x) along lon. One block per (b,c,lat) row.
// Layout: Xf_{re,im}[(lat*91 + f)*128 + (b*64+c)]  -> contiguous over bc.
// ---------------------------------------------------------------------------
__global__ void dft_x_kernel(const float* __restrict__ x,
                             float* __restrict__ Xf_re,
                             float* __restrict__ Xf_im) {
    const int row = blockIdx.x;          // (b*64+c)*91 + lat
    const int lat = row % NLAT;
    const int bc  = row / NLAT;
    __shared__ float sx[NLON], ctab[NLON], stab[NLON];
    const float TWOPI = 6.28318530717958647692f;
    for (int p = threadIdx.x; p < NLON; p += blockDim.x) {
        sx[p] = x[(size_t)row * NLON + p];
        float th = TWOPI * (float)p / (float)NLON;
        ctab[p] = cosf(th);
        stab[p] = sinf(th);
    }
    __syncthreads();
    for (int f = threadIdx.x; f < NFREQ; f += blockDim.x) {
        float sr = 0.f, si = 0.f;
        for (int p = 0; p < NLON; ++p) {
            int m = (f * p) % NLON;
            sr += sx[p] * ctab[m];
            si -= sx[p] * stab[m];        // e^{-i theta}
        }
        size_t o = ((size_t)lat * NFREQ + f) * BC + bc;
        Xf_re[o] = sr;
        Xf_im[o] = si;
    }
}

// ---------------------------------------------------------------------------
// Kernel 2: build psi on the fly (fp64 geometry, matches numpy precompute),
// DFT along p, store conj(Pf).
// One block per (t,w). Layout: Pc_{re,im}[((t*91+f)*25 + k)*9 + w]
// ---------------------------------------------------------------------------
__global__ void psi_dft_kernel(float* __restrict__ Pc_re,
                               float* __restrict__ Pc_im) {
    const int t = blockIdx.x / WWIN;
    const int w = blockIdx.x % WWIN;
    __shared__ float spsi[KSZ][NLON];
    __shared__ float ctab[NLON], stab[NLON];
    const double PI = 3.141592653589793238462643;
    const double cutoff = 4.0 * PI / 90.0;
    const double dr = cutoff / 4.0;             // NR-1 = 4
    const double dphi = 2.0 * PI / 6.0;         // NPHI = 6

    int ti_raw = t - 4 + w;
    bool valid = (ti_raw >= 0) && (ti_raw < NLAT);
    int ti = ti_raw < 0 ? 0 : (ti_raw > NLAT - 1 ? NLAT - 1 : ti_raw);
    double to  = PI * (double)t  / 90.0;
    double tia = PI * (double)ti / 90.0;
    double quad = sin(tia) * (PI / 90.0) * (2.0 * PI / 180.0);
    double cto = cos(to),  sto = sin(to);
    double cti = cos(tia), sti = sin(tia);

    for (int p = threadIdx.x; p < NLON; p += blockDim.x) {
        double ph = 2.0 * PI * (double)p / (double)NLON;
        ctab[p] = (float)cos(ph);
        stab[p] = (float)sin(ph);
        double cph = cos(ph), sph = sin(ph);
        double xx = cto * sti * cph - sto * cti;
        double yy = sti * sph;
        double zz = sto * sti * cph + cto * cti;
        zz = zz > 1.0 ? 1.0 : (zz < -1.0 ? -1.0 : zz);
        double r = acos(zz);
        double phi = atan2(yy, xx);
        if (phi < 0.0) phi += 2.0 * PI;
        bool inside = (r <= cutoff) && valid;
        double v0 = 1.0 - r / dr; if (v0 < 0.0) v0 = 0.0;
        spsi[0][p] = inside ? (float)(v0 * quad) : 0.f;
        for (int ir = 1; ir <= 4; ++ir) {
            double rad = 1.0 - fabs(r - (double)ir * dr) / dr;
            if (rad < 0.0) rad = 0.0;
            for (int ip = 0; ip < 6; ++ip) {
                double d = phi - (double)ip * dphi + PI;
                d = d - floor(d / (2.0 * PI)) * (2.0 * PI);
                double dp = fabs(d - PI);
                double ang = 1.0 - dp / dphi;
                if (ang < 0.0) ang = 0.0;
                spsi[1 + (ir - 1) * 6 + ip][p] =
                    inside ? (float)(rad * ang * quad) : 0.f;
            }
        }
    }
    __syncthreads();
    for (int i = threadIdx.x; i < KSZ * NFREQ; i += blockDim.x) {
        int k = i / NFREQ, f = i % NFREQ;
        float sr = 0.f, si = 0.f;
        for (int p = 0; p < NLON; ++p) {
            int m = (f * p) % NLON;
            float v = spsi[k][p];
            sr += v * ctab[m];
            si += v * stab[m];            // +sin: this IS conj(Pf)
        }
        size_t o = ((size_t)(t * NFREQ + f) * KSZ + k) * WWIN + w;
        Pc_re[o] = sr;
        Pc_im[o] = si;
    }
}

// ---------------------------------------------------------------------------
// Kernel 3 (stage A): Z[(c*25+k), n] = sum_w Pc(k,t,w,f) * Xf(b,c,ti(t,w),f)
// (complex product; Pc already conjugated). One block per (t,f).
// n = ((b*91+t)*91+f)*2 + {0:re,1:im};  Z row stride = NPAD.
// ---------------------------------------------------------------------------
__global__ void stageA_kernel(const float* __restrict__ Xf_re,
                              const float* __restrict__ Xf_im,
                              const float* __restrict__ Pc_re,
                              const float* __restrict__ Pc_im,
                              float* __restrict__ Z) {
    const int tf = blockIdx.x;           // t*91 + f
    const int t = tf / NFREQ, f = tf % NFREQ;
    __shared__ float sPr[KSZ * WWIN], sPi[KSZ * WWIN];
    __shared__ float sXr[WWIN][BC], sXi[WWIN][BC];
    const float* pr = &Pc_re[(size_t)tf * KSZ * WWIN];
    const float* pi = &Pc_im[(size_t)tf * KSZ * WWIN];
    for (int i = threadIdx.x; i < KSZ * WWIN; i += blockDim.x) {
        sPr[i] = pr[i]; sPi[i] = pi[i];
    }
    for (int i = threadIdx.x; i < WWIN * BC; i += blockDim.x) {
        int w = i >> 7, bc = i & (BC - 1);
        int ti = t - 4 + w;
        ti = ti < 0 ? 0 : (ti > NLAT - 1 ? NLAT - 1 : ti);
        size_t base = ((size_t)ti * NFREQ + f) * BC + bc;
        sXr[w][bc] = Xf_re[base];
        sXi[w][bc] = Xf_im[base];
    }
    __syncthreads();
    for (int i = threadIdx.x; i < NB * CIN * KSZ; i += blockDim.x) {
        int k = i % KSZ, rem = i / KSZ;
        int c = rem & (CIN - 1), b = rem >> 6;
        int bc = b * CIN + c;
        float zr = 0.f, zi = 0.f;
        for (int w = 0; w < WWIN; ++w) {
            float prw = sPr[k * WWIN + w], piw = sPi[k * WWIN + w];
            float xr = sXr[w][bc], xi = sXi[w][bc];
            zr += prw * xr - piw * xi;
            zi += prw * xi + piw * xr;
        }
        size_t n   = ((size_t)(b * NLAT + t) * NFREQ + f) * 2;
        size_t row = (size_t)(c * KSZ + k) * NPAD;
        Z[row + n]     = zr;
        Z[row + n + 1] = zi;
    }
}

// ---------------------------------------------------------------------------
// Kernel 4: WMMA fp32 GEMM. Of[o, n] = sum_ck W[o, ck] * Z[ck, n]
// M=64, K=1600, N=NPAD. One block (8 waves) per 16-col N tile; K split 8 ways
// across waves (200 each), partials reduced via LDS. Z (the 212 MB operand)
// is streamed from HBM with a depth-4 async global->LDS pipeline:
// one global_load_async_to_lds_b64 moves a full 4x16 B tile (256B) per wave.
// ---------------------------------------------------------------------------
__global__ void __launch_bounds__(256)
gemm_wmma_kernel(const float* __restrict__ W,
                 const float* __restrict__ Z,
                 float* __restrict__ Of) {
    __shared__ float stage[8][DEPTH][64];   // per-wave async staging (8 KB)
    __shared__ float red[8][1024];          // cross-wave reduction (32 KB)

    const int tid  = threadIdx.x;
    const int wv   = tid >> 5;              // wave id 0..7 -> K chunk
    const int lane = tid & 31;
    const int n0   = blockIdx.x * 16;       // grid == NTILES exactly
    const int lm   = lane & 15;
    const int lh   = lane >> 4;
    const int kb   = wv * KCHUNK;

    // async-copy lane mapping: tile[kk][nn], kk=lane>>3, nn=(lane&7)*2 (+1)
    const int srow = lane >> 3;
    const int scol = (lane & 7) * 2;
    const float* gsrc0 = Z + (size_t)(kb + srow) * NPAD + n0 + scol;
    const unsigned lds_lane = (unsigned)(srow * 16 + scol) * 4u;

    // prologue: enqueue DEPTH tiles
    #pragma unroll
    for (int j = 0; j < DEPTH; ++j) {
        unsigned lo = (unsigned)(uintptr_t)(&stage[wv][j][0]) + lds_lane;
        unsigned long long gp = (unsigned long long)(uintptr_t)(gsrc0 + (size_t)(4 * j) * NPAD);
        asm volatile("global_load_async_to_lds_b64 %0, %1, off"
                     :: "v"(lo), "v"(gp) : "memory");
    }

    v8f acc0 = {}, acc1 = {}, acc2 = {}, acc3 = {};
    auto compute = [&](int j) {
        const float* tile = &stage[wv][j & (DEPTH - 1)][0];
        v2f b;
        b.x = tile[(2 * lh) * 16 + lm];
        b.y = tile[(2 * lh + 1) * 16 + lm];
        const int ka = kb + 4 * j + 2 * lh;
        v2f a0 = *(const v2f*)&W[(size_t)( 0 + lm) * KDIM + ka];
        v2f a1 = *(const v2f*)&W[(size_t)(16 + lm) * KDIM + ka];
        v2f a2 = *(const v2f*)&W[(size_t)(32 + lm) * KDIM + ka];
        v2f a3 = *(const v2f*)&W[(size_t)(48 + lm) * KDIM + ka];
        acc0 = __builtin_amdgcn_wmma_f32_16x16x4_f32(false, a0, false, b,
                                                     (short)0, acc0, false, false);
        acc1 = __builtin_amdgcn_wmma_f32_16x16x4_f32(false, a1, false, b,
                                                     (short)0, acc1, false, false);
        acc2 = __builtin_amdgcn_wmma_f32_16x16x4_f32(false, a2, false, b,
                                                     (short)0, acc2, false, false);
        acc3 = __builtin_amdgcn_wmma_f32_16x16x4_f32(false, a3, false, b,
                                                     (short)0, acc3, false, false);
    };

    // main pipeline: wait oldest tile, consume, recycle slot with tile j+DEPTH
    for (int j = 0; j < KSTEPS - DEPTH; ++j) {
        asm volatile("s_wait_asynccnt 0x3" ::: "memory");   // tile j landed
        compute(j);
        asm volatile("s_wait_dscnt 0x0" ::: "memory");      // LDS reads drained
        unsigned lo = (unsigned)(uintptr_t)(&stage[wv][j & (DEPTH - 1)][0]) + lds_lane;
        unsigned long long gp =
            (unsigned long long)(uintptr_t)(gsrc0 + (size_t)(4 * (j + DEPTH)) * NPAD);
        asm volatile("global_load_async_to_lds_b64 %0, %1, off"
                     :: "v"(lo), "v"(gp) : "memory");
    }
    asm volatile("s_wait_asynccnt 0x0" ::: "memory");       // drain pipeline
    #pragma unroll
    for (int j = KSTEPS - DEPTH; j < KSTEPS; ++j) compute(j);

    // cross-wave K reduction through LDS
    #pragma unroll
    for (int jj = 0; jj < 8; ++jj) {
        red[wv][0 * 256 + jj * 32 + lane] = acc0[jj];
        red[wv][1 * 256 + jj * 32 + lane] = acc1[jj];
        red[wv][2 * 256 + jj * 32 + lane] = acc2[jj];
        red[wv][3 * 256 + jj * 32 + lane] = acc3[jj];
    }
    __syncthreads();
    for (int e = tid; e < 1024; e += 256) {
        float s = 0.f;
        #pragma unroll
        for (int ww = 0; ww < 8; ++ww) s += red[ww][e];
        const int a  = e >> 8;          // M tile
        const int jj = (e >> 5) & 7;    // accumulator VGPR index
        const int ln = e & 31;
        const int row = a * 16 + jj + 8 * (ln >> 4);
        const int col = n0 + (ln & 15);
        Of[(size_t)row * NPAD + col] = s;
    }
}

// ---------------------------------------------------------------------------
// Kernel 5: irfft (n=180). One block per (b,o,t).
// ---------------------------------------------------------------------------
__global__ void irfft_kernel(const float* __restrict__ Of,
                             float* __restrict__ out) {
    const int bot = blockIdx.x;          // (b*64+o)*91 + t
    const int t  = bot % NLAT;
    const int bo = bot / NLAT;
    const int o  = bo % COUT;
    const int b  = bo / COUT;
    __shared__ float fre[NFREQ], fim[NFREQ], ctab[NLON], stab[NLON];
    const float TWOPI = 6.28318530717958647692f;
    for (int p = threadIdx.x; p < NLON; p += blockDim.x) {
        float th = TWOPI * (float)p / (float)NLON;
        ctab[p] = cosf(th);
        stab[p] = sinf(th);
    }
    for (int f = threadIdx.x; f < NFREQ; f += blockDim.x) {
        size_t n = ((size_t)(b * NLAT + t) * NFREQ + f) * 2;
        fre[f] = Of[(size_t)o * NPAD + n];
        fim[f] = Of[(size_t)o * NPAD + n + 1];
    }
    __syncthreads();
    for (int p = threadIdx.x; p < NLON; p += blockDim.x) {
        float s = fre[0];
        for (int f = 1; f < 90; ++f) {
            int m = (f * p) % NLON;
            s += 2.f * (fre[f] * ctab[m] - fim[f] * stab[m]);
        }
        s += (p & 1) ? -fre[90] : fre[90];   // Nyquist, cos(pi*p)
        out[(size_t)bot * NLON + p] = s * (1.0f / 180.0f);
    }
}

// ---------------------------------------------------------------------------
extern "C" void kernel_launch(void* const* d_in, const int* in_sizes, int n_in,
                              void* d_out, int out_size, void* d_ws, size_t ws_size,
                              hipStream_t stream) {
    const float* x = (const float*)d_in[0];       // (2,64,91,180) f32
    const float* w = (const float*)d_in[1];       // (64,64,25) f32
    float* out = (float*)d_out;                   // (2,64,91,180) f32

    auto alignup = [](size_t v) { return (v + 255) & ~(size_t)255; };
    char* base = (char*)d_ws;
    size_t off = 0;
    const size_t szXf = alignup((size_t)NLAT * NFREQ * BC * sizeof(float));
    const size_t szPc = alignup((size_t)NLAT * NFREQ * KSZ * WWIN * sizeof(float));
    const size_t szZ  = alignup((size_t)KDIM * NPAD * sizeof(float));
    const size_t szOf = alignup((size_t)COUT * NPAD * sizeof(float));
    float* Xf_re = (float*)(base + off); off += szXf;
    float* Xf_im = (float*)(base + off); off += szXf;
    float* Pc_re = (float*)(base + off); off += szPc;
    float* Pc_im = (float*)(base + off); off += szPc;
    float* Z     = (float*)(base + off); off += szZ;
    float* Of    = (float*)(base + off); off += szOf;
    (void)ws_size; (void)in_sizes; (void)n_in; (void)out_size;

    dft_x_kernel<<<NB * CIN * NLAT, 192, 0, stream>>>(x, Xf_re, Xf_im);
    psi_dft_kernel<<<NLAT * WWIN, 192, 0, stream>>>(Pc_re, Pc_im);
    stageA_kernel<<<NLAT * NFREQ, 256, 0, stream>>>(Xf_re, Xf_im, Pc_re, Pc_im, Z);
    gemm_wmma_kernel<<<NTILES, 256, 0, stream>>>(w, Z, Of);
    irfft_kernel<<<NB * COUT * NLAT, 192, 0, stream>>>(Of, out);
}